// PureTriadicBrain_73753178407487
// MI455X (gfx1250) — compile-verified
//
#include <hip/hip_runtime.h>
#include <math.h>

// ---- problem constants (from reference) ----
#define BATCH 256
#define NN 64          // nodes
#define DD 64          // state dim
#define HH 16          // heads = D/4
#define HEADD 4        // head dim
#define UNTANH 0.6f
#define INV_BSCALE 0.5411961001f   // 1/sqrt(2+sqrt(2))
#define INV_SQRT_HEAD 0.5f         // 1/sqrt(4)
#define SINK_ITERS 10
#define SINK_EPS 1e-8f
#define NELEM (BATCH*NN*DD)        // 1048576

typedef __attribute__((ext_vector_type(2))) float v2f;
typedef __attribute__((ext_vector_type(8))) float v8f;
typedef __attribute__((ext_vector_type(4))) unsigned int v4u;
typedef __attribute__((ext_vector_type(8))) int v8i;
typedef __attribute__((ext_vector_type(4))) int v4i;

__device__ __forceinline__ float buntanh(float x) {
    return (tanhf(x) + UNTANH * x) * INV_BSCALE;
}

// -------------------------------------------------------------------------
// Kernel A: per-(b,n,which) GEMV fused with buntanh.
// The 805MB weight stream is the roofline (~35us @ 23.3TB/s), so feed it
// through the Tensor Data Mover: one wave per 64x64 fp32 matrix, one
// tensor_load_to_lds DMA per matrix, s_wait_tensorcnt, then conflict-free
// b64 LDS reads for the dot products.  ~20 concurrent 16KB tiles per WGP
// keep HBM busy.  grid = 3*B*N blocks, 32 threads (exactly one wave so the
// EXEC-ignoring tensor op is issued exactly once).
// -------------------------------------------------------------------------
__global__ void __launch_bounds__(32)
matvec_buntanh_kernel(const float* __restrict__ state,
                      const float* __restrict__ w1,
                      const float* __restrict__ w2,
                      const float* __restrict__ w3,
                      float* __restrict__ pred,
                      float* __restrict__ keys,
                      float* __restrict__ que) {
    __shared__ float smem[DD * DD + DD];       // [0,4096): W tile, [4096,4160): x
    float* __restrict__ Ws  = smem;
    float* __restrict__ xsm = smem + DD * DD;

    const int which = blockIdx.x % 3;          // 0:w1 1:w2 2:w3
    const int bn    = blockIdx.x / 3;          // 0 .. B*N-1
    const int l     = threadIdx.x;             // lane 0..31

    const float* __restrict__ W =
        (which == 0 ? w1 : which == 1 ? w2 : w3) + (size_t)bn * (DD * DD);
    float* __restrict__ outp =
        (which == 0 ? pred : which == 1 ? keys : que) + (size_t)bn * DD;

    // stage x row into LDS (2 elements per lane)
    xsm[l]      = state[bn * DD + l];
    xsm[l + 32] = state[bn * DD + l + 32];

    // ---- Tensor DMA descriptor (D#): 64x64 fp32 tile -> LDS offset 0 ----
    {
        const unsigned long long ga = (unsigned long long)(uintptr_t)W;
        // group0: count=1 | lds_addr | global_addr[56:0] | type=2
        v4u g0;
        g0.x = 1u;                                     // count=1, user mode
        g0.y = 0u;                                     // lds_addr = 0
        g0.z = (unsigned)(ga & 0xffffffffu);           // global_addr[31:0]
        g0.w = (unsigned)((ga >> 32) & 0x01ffffffu)    // global_addr[56:32]
               | 0x80000000u;                          // type=2 ("image")
        // group1: data_size=4B, tensor 64x64, tile 64x64, dim0_stride=64
        v8i g1;
        g1[0] = (int)(2u << 16);     // workgroup_mask=0, data_size=2 (4B)
        g1[1] = (int)(64u << 16);    // tensor_dim0=64      (bits 79:48)
        g1[2] = (int)(64u << 16);    // tensor_dim1=64      (bits 111:80)
        g1[3] = (int)(64u << 16);    // tile_dim0=64        (bits 127:112)
        g1[4] = 64;                  // tile_dim1=64        (bits 143:128)
        g1[5] = 64;                  // tensor_dim0_stride  (bits 207:160)
        g1[6] = 0;
        g1[7] = 0;
        v4i g2 = {0, 0, 0, 0};       // no dims 2..4
        v4i g3 = {0, 0, 0, 0};
        v8i g4 = {0, 0, 0, 0, 0, 0, 0, 0};   // extra group (6-arg builtin form)
        __builtin_amdgcn_tensor_load_to_lds(g0, g1, g2, g3, g4, 0);
    }
    __builtin_amdgcn_s_wait_tensorcnt(0);
    __syncthreads();                 // make xsm stores + DMA visible wave-wide

    // ---- GEMV from LDS: lane l owns columns {2l, 2l+1} ----
    float a0 = 0.f, a1 = 0.f;
#pragma unroll 8
    for (int j = 0; j < DD; ++j) {
        const float xj = xsm[j];
        const float2 wv = *(const float2*)(Ws + j * DD + 2 * l);
        a0 = fmaf(xj, wv.x, a0);
        a1 = fmaf(xj, wv.y, a1);
    }
    float2 o; o.x = buntanh(a0); o.y = buntanh(a1);
    *(float2*)(outp + 2 * l) = o;    // 256B coalesced store
}

// -------------------------------------------------------------------------
// Kernel B: per-(b,h) attention + sinkhorn + routed_V.
// QK^T (64x64, K=4) done with V_WMMA_F32_16X16X4_F32: 2 waves x 8 tiles.
// grid = B*H blocks, 64 threads (2 full waves; EXEC all ones for WMMA).
// -------------------------------------------------------------------------
__global__ void __launch_bounds__(64)
attention_sinkhorn_kernel(const float* __restrict__ que,
                          const float* __restrict__ keys,
                          const float* __restrict__ pred,
                          float* __restrict__ tgt) {
    const int b = blockIdx.x >> 4;
    const int h = blockIdx.x & 15;
    const int n = threadIdx.x;           // 0..63 : row this thread owns

    __shared__ float Qs[NN * HEADD];     // 64x4
    __shared__ float Ks[NN * HEADD];
    __shared__ float Vs[NN * HEADD];
    __shared__ float As[NN * NN];        // 64x64, reused as P
    __shared__ float red_ss[NN];
    __shared__ float red_mx[NN];

    // gather head slices (16B aligned float4 loads)
    {
        const size_t base = ((size_t)(b * NN + n)) * DD + h * HEADD;
        const float4 q4 = *(const float4*)(que  + base);
        const float4 k4 = *(const float4*)(keys + base);
        const float4 v4 = *(const float4*)(pred + base);
        *(float4*)(Qs + n * HEADD) = q4;
        *(float4*)(Ks + n * HEADD) = k4;
        *(float4*)(Vs + n * HEADD) = v4;
    }
    __syncthreads();

    // ---- raw_A = Q K^T / sqrt(hd) via WMMA f32 16x16x4 ----
    // wave w computes tile rows {2w, 2w+1} x tile cols {0..3}
    {
        const int wv    = threadIdx.x >> 5;   // wave id 0/1
        const int lane  = threadIdx.x & 31;
        const int lhalf = lane >> 4;          // lane half 0/1
        const int l15   = lane & 15;
        const int k0    = lhalf * 2;          // K-pair this lane holds
#pragma unroll
        for (int t = 0; t < 8; ++t) {
            const int ti = wv * 2 + (t >> 2); // tile row 0..3
            const int tj = t & 3;             // tile col 0..3
            // A fragment: rows of Q (16x4): lanes 0-15 K={0,1}, 16-31 K={2,3}
            const int mrow = ti * 16 + l15;
            v2f af; af.x = Qs[mrow * HEADD + k0];
                    af.y = Qs[mrow * HEADD + k0 + 1];
            // B fragment: K^T (4x16): element (k, ncol) = K[ncol][k]
            const int ncol = tj * 16 + l15;
            v2f bf; bf.x = Ks[ncol * HEADD + k0];
                    bf.y = Ks[ncol * HEADD + k0 + 1];
            v8f c = {};
            c = __builtin_amdgcn_wmma_f32_16x16x4_f32(
                    /*neg_a=*/false, af, /*neg_b=*/false, bf,
                    /*c_mod=*/(short)0, c, /*reuse_a=*/false, /*reuse_b=*/false);
            // C layout: VGPR r -> row r (lanes 0-15) / row r+8 (lanes 16-31)
#pragma unroll
            for (int r = 0; r < 8; ++r) {
                const int mm = ti * 16 + r + lhalf * 8;
                As[mm * NN + tj * 16 + l15] = c[r] * INV_SQRT_HEAD;
            }
        }
    }
    __syncthreads();

    // ---- rms_norm over the whole 64x64 matrix + matrix max ----
    float ss = 0.f, mx = -3.4e38f;
    for (int m = 0; m < NN; ++m) {
        const float v = As[n * NN + m];
        ss += v * v;
        mx = fmaxf(mx, v);
    }
    red_ss[n] = ss;
    red_mx[n] = mx;
    __syncthreads();
    float tot_ss = 0.f, tot_mx = -3.4e38f;
    for (int i = 0; i < NN; ++i) {       // broadcast reads, fixed order
        tot_ss += red_ss[i];
        tot_mx = fmaxf(tot_mx, red_mx[i]);
    }
    const float scale = rsqrtf(tot_ss * (1.0f / (NN * NN)) + 1e-6f);
    const float gmax  = tot_mx * scale;  // max of normalized matrix (scale>0)

    // ---- p = exp(x - max), in place ----
    for (int m = 0; m < NN; ++m)
        As[n * NN + m] = expf(As[n * NN + m] * scale - gmax);
    __syncthreads();

    // ---- Sinkhorn-Knopp: 10 x (row norm, col norm) ----
    for (int it = 0; it < SINK_ITERS; ++it) {
        float rs = 0.f;                   // thread n owns row n
        for (int m = 0; m < NN; ++m) rs += As[n * NN + m];
        const float rinv = 1.0f / (rs + SINK_EPS);
        for (int m = 0; m < NN; ++m) As[n * NN + m] *= rinv;
        __syncthreads();
        float cs = 0.f;                   // thread n owns column n (bank-clean)
        for (int m = 0; m < NN; ++m) cs += As[m * NN + n];
        const float cinv = 1.0f / (cs + SINK_EPS);
        for (int m = 0; m < NN; ++m) As[m * NN + n] *= cinv;
        __syncthreads();
    }

    // ---- k-hot threshold + routed_V = A @ V (64x64 @ 64x4) ----
    float acc0 = 0.f, acc1 = 0.f, acc2 = 0.f, acc3 = 0.f;
    for (int m = 0; m < NN; ++m) {
        float p = As[n * NN + m];
        p = (p > (1.0f / NN)) ? p : 0.0f;
        acc0 = fmaf(p, Vs[m * HEADD + 0], acc0);
        acc1 = fmaf(p, Vs[m * HEADD + 1], acc1);
        acc2 = fmaf(p, Vs[m * HEADD + 2], acc2);
        acc3 = fmaf(p, Vs[m * HEADD + 3], acc3);
    }
    float4 o; o.x = acc0; o.y = acc1; o.z = acc2; o.w = acc3;
    *(float4*)(tgt + ((size_t)(b * NN + n)) * DD + h * HEADD) = o;
}

// -------------------------------------------------------------------------
// Kernel C: partial sums of the three MSE losses (env-pad override for n<2).
// -------------------------------------------------------------------------
__global__ void __launch_bounds__(256)
loss_partial_kernel(const float* __restrict__ pred,
                    const float* __restrict__ keys,
                    const float* __restrict__ que,
                    const float* __restrict__ tgt,
                    const float* __restrict__ state,
                    const float* __restrict__ env,
                    float* __restrict__ partials) {
    const int tid = threadIdx.x;
    float acc = 0.f;
    for (int idx = blockIdx.x * 256 + tid; idx < NELEM; idx += gridDim.x * 256) {
        const int b = idx / (NN * DD);
        const int r = idx - b * (NN * DD);
        const int nn = r / DD;
        const int kk = r - nn * DD;
        float t;
        if (nn < 2) {                         // env pad rows (ceil(100/64)=2)
            const int e = nn * DD + kk;
            t = (e < 100) ? env[b * 100 + e] : 0.0f;
        } else {
            t = tgt[idx];
        }
        const float d1 = pred[idx] - t;
        const float d2 = keys[idx] - state[idx];
        const float d3 = que[idx]  - t;
        acc += d1 * d1 + d2 * d2 + d3 * d3;
    }
    __shared__ float sm[256];
    sm[tid] = acc;
    __syncthreads();
#pragma unroll
    for (int s = 128; s > 0; s >>= 1) {
        if (tid < s) sm[tid] += sm[tid + s];
        __syncthreads();
    }
    if (tid == 0) partials[blockIdx.x] = sm[0];
}

__global__ void __launch_bounds__(256)
loss_final_kernel(const float* __restrict__ partials, float* __restrict__ out) {
    const int tid = threadIdx.x;
    float acc = 0.f;
    for (int i = tid; i < 1024; i += 256) acc += partials[i];
    __shared__ float sm[256];
    sm[tid] = acc;
    __syncthreads();
#pragma unroll
    for (int s = 128; s > 0; s >>= 1) {
        if (tid < s) sm[tid] += sm[tid + s];
        __syncthreads();
    }
    if (tid == 0) out[0] = sm[0] * (1.0f / (float)NELEM);
}

// -------------------------------------------------------------------------
extern "C" void kernel_launch(void* const* d_in, const int* in_sizes, int n_in,
                              void* d_out, int out_size, void* d_ws, size_t ws_size,
                              hipStream_t stream) {
    (void)in_sizes; (void)n_in; (void)out_size; (void)ws_size;
    const float* env   = (const float*)d_in[0];  // [B,100]
    const float* state = (const float*)d_in[1];  // [B,N,D]
    const float* w1    = (const float*)d_in[2];  // [B,N,D,D]
    const float* w2    = (const float*)d_in[3];
    const float* w3    = (const float*)d_in[4];

    float* ws       = (float*)d_ws;
    float* pred     = ws;                 // 1M floats
    float* keys     = ws + 1 * NELEM;
    float* que      = ws + 2 * NELEM;
    float* tgt      = ws + 3 * NELEM;
    float* partials = ws + 4 * NELEM;     // 1024 floats

    matvec_buntanh_kernel<<<3 * BATCH * NN, 32, 0, stream>>>(state, w1, w2, w3,
                                                             pred, keys, que);
    attention_sinkhorn_kernel<<<BATCH * HH, 64, 0, stream>>>(que, keys, pred, tgt);
    loss_partial_kernel<<<1024, 256, 0, stream>>>(pred, keys, que, tgt, state,
                                                  env, partials);
    loss_final_kernel<<<1, 256, 0, stream>>>(partials, (float*)d_out);
}